// GraphConvolutionLayer_63058709840593
// MI455X (gfx1250) — compile-verified
//
#include <hip/hip_runtime.h>
#include <hip/hip_bf16.h>

typedef __attribute__((ext_vector_type(2))) float v2f;
typedef __attribute__((ext_vector_type(8))) float v8f;

#define IN_F  128
#define OUT_F 128

// ---------------------------------------------------------------------------
// Kernel 0: zero the neighborhood accumulator (float4 stores, grid-stride)
// ---------------------------------------------------------------------------
__global__ void gcn_zero_kernel(float4* __restrict__ p, long n4) {
    long tid = (long)blockIdx.x * blockDim.x + threadIdx.x;
    long stride = (long)gridDim.x * blockDim.x;
    float4 z = make_float4(0.f, 0.f, 0.f, 0.f);
    for (long i = tid; i < n4; i += stride) p[i] = z;
}

// ---------------------------------------------------------------------------
// Kernel 1: edge scatter  neigh[row[e]] += val[e] * X[col[e]]
// One wave32 per edge; each lane handles 4 consecutive features (float4 gather
// + 4 global f32 atomic adds). Entire working set is L2-resident (~51 MB).
// ---------------------------------------------------------------------------
__global__ void gcn_scatter_kernel(const int*   __restrict__ adj_row,
                                   const int*   __restrict__ adj_col,
                                   const float* __restrict__ adj_val,
                                   const float* __restrict__ x,
                                   float*       __restrict__ neigh,
                                   int n_edges) {
    long tid    = (long)blockIdx.x * blockDim.x + threadIdx.x;
    long stride = (long)gridDim.x * blockDim.x;
    long total  = (long)n_edges * 32;
    for (; tid < total; tid += stride) {
        int e    = (int)(tid >> 5);
        int lane = (int)(tid & 31);
        int r = adj_row[e];
        int c = adj_col[e];
        float v = adj_val[e];
        const float4 xin = *(const float4*)(x + (long)c * IN_F + lane * 4);
        float* dst = neigh + (long)r * IN_F + lane * 4;
        atomicAdd(dst + 0, v * xin.x);
        atomicAdd(dst + 1, v * xin.y);
        atomicAdd(dst + 2, v * xin.z);
        atomicAdd(dst + 3, v * xin.w);
    }
}

// ---------------------------------------------------------------------------
// Kernel 2: out = relu(neigh @ W + b) via V_WMMA_F32_16X16X4_F32.
// 256 threads = 8 waves; wave w owns the 16-column panel n0 = 16*w.
// Each wave preloads its B panel (K=128 -> 32 k-steps -> 64 VGPRs) once,
// then grid-strides over 16-row M tiles, doing 32 WMMAs per tile.
// ---------------------------------------------------------------------------
__global__ __launch_bounds__(256)
void gcn_gemm_bias_relu_kernel(const float* __restrict__ neigh,
                               const float* __restrict__ W,
                               const float* __restrict__ bias,
                               float*       __restrict__ out,
                               int n_rows) {
    const int lane    = threadIdx.x & 31;
    const int wave    = threadIdx.x >> 5;      // 0..7 -> column panel
    const int n0      = wave * 16;
    const int halfSel = lane >> 4;             // 0: lanes 0-15, 1: lanes 16-31
    const int lidx    = lane & 15;
    const int n       = n0 + lidx;

    // --- Preload B fragments: B[k][n] -> vgpr (k&1), lane n + 16*(k>>1) ---
    v2f bfrag[32];
#pragma unroll
    for (int ks = 0; ks < 32; ++ks) {
        const int kk = ks * 4 + 2 * halfSel;
        bfrag[ks].x = W[(long)kk       * OUT_F + n];
        bfrag[ks].y = W[(long)(kk + 1) * OUT_F + n];
    }
    const float bv = bias[n];

    const int numTilesM = n_rows >> 4;   // 50000/16 = 3125
    for (int tm = blockIdx.x; tm < numTilesM; tm += gridDim.x) {
        const int m0 = tm * 16;
        v8f c = {0.f, 0.f, 0.f, 0.f, 0.f, 0.f, 0.f, 0.f};

        // A[m][k]: lane = (m-m0) + 16*((k%4)>>1), vgpr = k&1
        const float* arow = neigh + (long)(m0 + lidx) * IN_F + 2 * halfSel;
#pragma unroll
        for (int ks = 0; ks < 32; ++ks) {
            v2f a;
            a.x = arow[ks * 4 + 0];
            a.y = arow[ks * 4 + 1];
            c = __builtin_amdgcn_wmma_f32_16x16x4_f32(
                    /*neg_a=*/false, a, /*neg_b=*/false, bfrag[ks],
                    /*c_mod=*/(short)0, c, /*reuse_a=*/false, /*reuse_b=*/false);
        }

        // D[r + 8*halfSel][n] = c[r]  -> bias + relu + store
#pragma unroll
        for (int r = 0; r < 8; ++r) {
            const int m = m0 + r + 8 * halfSel;
            float vo = c[r] + bv;
            out[(long)m * OUT_F + n] = vo > 0.f ? vo : 0.f;
        }
    }
}

// ---------------------------------------------------------------------------
// Host launcher
// Inputs (setup_inputs order): adj_row[i32], adj_col[i32], adj_val[f32],
//                              inputs[f32 N*128], weight[f32 128*128], b[f32 128]
// Output: f32 [N, 128].  Workspace: neigh accumulator, N*128*4 bytes.
// ---------------------------------------------------------------------------
extern "C" void kernel_launch(void* const* d_in, const int* in_sizes, int n_in,
                              void* d_out, int out_size, void* d_ws, size_t ws_size,
                              hipStream_t stream) {
    const int*   adj_row = (const int*)  d_in[0];
    const int*   adj_col = (const int*)  d_in[1];
    const float* adj_val = (const float*)d_in[2];
    const float* x       = (const float*)d_in[3];
    const float* W       = (const float*)d_in[4];
    const float* bias    = (const float*)d_in[5];
    float*       out     = (float*)d_out;
    float*       neigh   = (float*)d_ws;

    const int n_edges = in_sizes[0];
    const int n_nodes = in_sizes[3] / IN_F;

    // 1) zero the accumulator
    long n4 = (long)n_nodes * IN_F / 4;
    gcn_zero_kernel<<<4096, 256, 0, stream>>>((float4*)neigh, n4);

    // 2) scatter-add edges (one wave per edge)
    long total_thr = (long)n_edges * 32;
    int  sblocks   = (int)((total_thr + 255) / 256);
    gcn_scatter_kernel<<<sblocks, 256, 0, stream>>>(adj_row, adj_col, adj_val,
                                                    x, neigh, n_edges);

    // 3) WMMA GEMM + bias + relu
    gcn_gemm_bias_relu_kernel<<<640, 256, 0, stream>>>(neigh, W, bias, out, n_nodes);
}